// VQEModel_1236950581428
// MI455X (gfx1250) — compile-verified
//
#include <hip/hip_runtime.h>
#include <math.h>

// CDNA5 wave32 WMMA vector types
typedef __attribute__((ext_vector_type(2))) float v2f;   // A/B operand of wmma_f32_16x16x4_f32
typedef __attribute__((ext_vector_type(8))) float v8f;   // C/D accumulator (16x16 f32)

#define NQ   8
#define QDIM 256

__device__ __forceinline__ float2 cmul(float2 a, float2 b) {
  return make_float2(a.x * b.x - a.y * b.y, a.x * b.y + a.y * b.x);
}
__device__ __forceinline__ float2 cadd(float2 a, float2 b) {
  return make_float2(a.x + b.x, a.y + b.y);
}

// ---------------------------------------------------------------------------
// Kernel 1: build the batch-invariant circuit unitary W (Rot layers + CNOT
// rings only; the per-sample RX embedding is NOT part of W).
// Thread j simulates basis column e_j. Column lives in LDS [amp][lane]
// (stride 32 float2 -> bank-conflict-free b64 access).
// Output layout (for WMMA B-operand b64 loads in kernel 2):
//   wR[((k>>1)*256 + n)*2 + (k&1)] = Re Wt[k][n],  Wt[k][n] = W[n][k]
// so a single float2 load at (k2*256+n) yields {Wt[2k2,n], Wt[2k2+1,n]}.
// ---------------------------------------------------------------------------
__global__ __launch_bounds__(32) void build_W(const float* __restrict__ weights,
                                              int n_layers,
                                              float* __restrict__ wR,
                                              float* __restrict__ wI) {
  __shared__ float2 col[QDIM][32];   // 64 KB
  const int t = threadIdx.x;
  const int j = blockIdx.x * 32 + t;

  for (int i = 0; i < QDIM; ++i) col[i][t] = make_float2(i == j ? 1.f : 0.f, 0.f);

  for (int l = 0; l < n_layers; ++l) {
    // --- 8 Rot gates (PennyLane Rot = RZ(om) RY(th) RZ(ph)) ---
    for (int q = 0; q < NQ; ++q) {
      const float ph = weights[(l * NQ + q) * 3 + 0];
      const float th = weights[(l * NQ + q) * 3 + 1];
      const float om = weights[(l * NQ + q) * 3 + 2];
      const float c  = cosf(0.5f * th), s = sinf(0.5f * th);
      const float ap = 0.5f * (ph + om), bm = 0.5f * (ph - om);
      const float2 u00 = make_float2( cosf(ap) * c, -sinf(ap) * c);  // e^{-i ap} c
      const float2 u01 = make_float2(-cosf(bm) * s, -sinf(bm) * s);  // -e^{+i bm} s
      const float2 u10 = make_float2( cosf(bm) * s, -sinf(bm) * s);  // e^{-i bm} s
      const float2 u11 = make_float2( cosf(ap) * c,  sinf(ap) * c);  // e^{+i ap} c
      const int stride = 1 << (7 - q);   // wire q <-> bit (7-q)
      for (int m = 0; m < QDIM / 2; ++m) {
        const int low = m & (stride - 1);
        const int hi  = m >> (7 - q);
        const int i0  = hi * (stride << 1) + low;
        const int i1  = i0 + stride;
        const float2 a0 = col[i0][t], a1 = col[i1][t];
        col[i0][t] = cadd(cmul(u00, a0), cmul(u01, a1));
        col[i1][t] = cadd(cmul(u10, a0), cmul(u11, a1));
      }
    }
    // --- CNOT ring, range r = l%7+1, applied in q order (order matters) ---
    const int r = l % (NQ - 1) + 1;
    for (int q = 0; q < NQ; ++q) {
      const int tq    = (q + r) & 7;
      const int cmask = 1 << (7 - q);
      const int tmask = 1 << (7 - tq);
      for (int m = 0; m < QDIM; ++m) {      // uniform across lanes: no divergence
        if ((m & cmask) && !(m & tmask)) {
          const int m2 = m | tmask;
          const float2 tmp = col[m][t];
          col[m][t]  = col[m2][t];
          col[m2][t] = tmp;
        }
      }
    }
  }

  // store W^T in paired-K layout (k == j for this thread)
  for (int n = 0; n < QDIM; ++n) {
    const int idx = (((j >> 1) * QDIM) + n) * 2 + (j & 1);
    wR[idx] = col[n][t].x;
    wI[idx] = col[n][t].y;
  }
}

// ---------------------------------------------------------------------------
// Kernel 2: fused embedding + complex GEMM (WMMA f32 16x16x4) + readout.
// One wave (32 threads) per block, 16 samples per wave, 2048 blocks.
//
// WMMA f32 16x16x4 layouts (wave32):
//   A (16x4):  lane l -> M = l&15; VGPR0 = K0 (l<16) / K2 (l>=16); VGPR1 = K1/K3
//   B (4x16):  lane l -> N = l&15; VGPR0 = K0 / K2;                VGPR1 = K1/K3
//   C/D:       VGPR r, lane l -> M = r + 8*(l>=16), N = l&15
// A comes from LDS (Phi, row stride 260 -> conflict-free ds_load_b64),
// B comes from L2-resident paired-K W^T via global_load_b64.
// f32 WMMA has no A/B NEG bits, so the complex GEMM uses 4 accumulators:
//   Psi_r = C1 - C2,  Psi_i = C3 + C4.
// ---------------------------------------------------------------------------
__global__ __launch_bounds__(32) void vqe_wmma(const float* __restrict__ x,
                                               const float* __restrict__ fcw,
                                               const float* __restrict__ fcb,
                                               const float* __restrict__ wR,
                                               const float* __restrict__ wI,
                                               float* __restrict__ out) {
  __shared__ float PhiR[16][260];   // padded stride: banks (4*s + k) % 64 distinct
  __shared__ float PhiI[16][260];
  __shared__ float zred[16][20];    // per-sample partial sums, 16B-aligned rows

  const int l = threadIdx.x;
  const int h = l >> 4;       // half-wave
  const int c = l & 15;       // sample-in-tile (for A/M) and column class (for B/N)
  const int sBase = blockIdx.x * 16;

  // --- embedding trig for this lane's sample ---
  float cs[NQ], sn[NQ];
  {
    const float* xp = x + (size_t)(sBase + c) * NQ;
#pragma unroll
    for (int q = 0; q < NQ; ++q) {
      const float xv = 0.5f * xp[q];
      cs[q] = cosf(xv);
      sn[q] = sinf(xv);
    }
  }

  // --- fill Phi: amplitude(b) = (prod bit? sin:cos) * (-i)^popcount(b) ---
  // lane handles sample c, amplitudes b in [h*128, h*128+128)
  for (int tt = 0; tt < 128; ++tt) {
    const int b = (h << 7) | tt;
    float m = h ? sn[0] : cs[0];           // qubit 0 = MSB
#pragma unroll
    for (int q = 1; q < NQ; ++q)
      m *= ((b >> (7 - q)) & 1) ? sn[q] : cs[q];
    const int p = __popc(b) & 3;           // (-i)^p: 0->(1,0) 1->(0,-1) 2->(-1,0) 3->(0,1)
    PhiR[c][b] = (p == 0) ? m : ((p == 2) ? -m : 0.f);
    PhiI[c][b] = (p == 1) ? -m : ((p == 3) ? m : 0.f);
  }
  __syncthreads();

  float fw[NQ];
#pragma unroll
  for (int i = 0; i < NQ; ++i) fw[i] = fcw[i];

  float zacc[8];
#pragma unroll
  for (int r = 0; r < 8; ++r) zacc[r] = 0.f;

  const float* aRbase = &PhiR[c][2 * h];
  const float* aIbase = &PhiI[c][2 * h];

  for (int nt = 0; nt < 16; ++nt) {
    const int n = nt * 16 + c;
    // g_n = sum_i fc_w[i] * (1 - 2*bit_{7-i}(n))  (folds Z-expectations + FC)
    float g = 0.f;
#pragma unroll
    for (int i = 0; i < NQ; ++i)
      g += fw[i] * (1.f - 2.f * (float)((n >> (7 - i)) & 1));

    v8f cr1 = {0.f, 0.f, 0.f, 0.f, 0.f, 0.f, 0.f, 0.f};
    v8f cr2 = cr1, ci1 = cr1, ci2 = cr1;

    const float* bR = wR + (size_t)(h * 512 + 2 * n);
    const float* bI = wI + (size_t)(h * 512 + 2 * n);

#pragma unroll 4
    for (int k0 = 0; k0 < QDIM; k0 += 4) {
      const v2f ar = *(const v2f*)(aRbase + k0);                     // ds_load_b64
      const v2f ai = *(const v2f*)(aIbase + k0);
      const v2f br = *(const v2f*)(bR + (size_t)(k0 >> 1) * 512);    // global_load_b64
      const v2f bi = *(const v2f*)(bI + (size_t)(k0 >> 1) * 512);
      cr1 = __builtin_amdgcn_wmma_f32_16x16x4_f32(false, ar, false, br, (short)0, cr1, false, false);
      cr2 = __builtin_amdgcn_wmma_f32_16x16x4_f32(false, ai, false, bi, (short)0, cr2, false, false);
      ci1 = __builtin_amdgcn_wmma_f32_16x16x4_f32(false, ar, false, bi, (short)0, ci1, false, false);
      ci2 = __builtin_amdgcn_wmma_f32_16x16x4_f32(false, ai, false, br, (short)0, ci2, false, false);
    }

#pragma unroll
    for (int r = 0; r < 8; ++r) {
      const float pr = cr1[r] - cr2[r];    // Re(Psi)
      const float pi = ci1[r] + ci2[r];    // Im(Psi)
      zacc[r] += (pr * pr + pi * pi) * g;  // |Psi|^2 weighted readout
    }
  }

  // C layout: zacc[r] on lane l belongs to sample (r + 8h), column class c
#pragma unroll
  for (int r = 0; r < 8; ++r) zred[r + 8 * h][c] = zacc[r];
  __syncthreads();

  if (l < 16) {
    float sum = fcb[0];
#pragma unroll
    for (int cc = 0; cc < 16; ++cc) sum += zred[l][cc];
    out[sBase + l] = 1.f / (1.f + expf(-sum));   // sigmoid
  }
}

// ---------------------------------------------------------------------------
extern "C" void kernel_launch(void* const* d_in, const int* in_sizes, int n_in,
                              void* d_out, int out_size, void* d_ws, size_t ws_size,
                              hipStream_t stream) {
  const float* x       = (const float*)d_in[0];   // (B, 8)
  const float* weights = (const float*)d_in[1];   // (L, 8, 3)
  const float* fcw     = (const float*)d_in[2];   // (1, 8)
  const float* fcb     = (const float*)d_in[3];   // (1,)
  float* out = (float*)d_out;

  // workspace: W^T real / imag in paired-K layout, 256 KB each
  float* wR = (float*)d_ws;
  float* wI = wR + QDIM * QDIM;

  const int B        = in_sizes[0] / NQ;
  const int n_layers = in_sizes[1] / (NQ * 3);

  build_W<<<QDIM / 32, 32, 0, stream>>>(weights, n_layers, wR, wI);
  vqe_wmma<<<B / 16, 32, 0, stream>>>(x, fcw, fcb, wR, wI, out);
}